// Quantizer_86535001080174
// MI455X (gfx1250) — compile-verified
//
#include <hip/hip_runtime.h>

typedef __attribute__((ext_vector_type(2))) float v2f;
typedef __attribute__((ext_vector_type(8))) float v8f;

#define N_ROWS 131072
#define K_CODES 1024
#define DIM 10
#define DPAD 12
#define WAVES_PER_BLOCK 8
#define ROWS_PER_WAVE 16
#define ROWS_PER_BLOCK (WAVES_PER_BLOCK * ROWS_PER_WAVE) /* 128 */
#define NUM_BLOCKS (N_ROWS / ROWS_PER_BLOCK)             /* 1024 */
#define TILES (K_CODES / 16)                             /* 64  */

struct Frag {
    v2f b0, b1, b2;
    float es;
};

__device__ __forceinline__ Frag load_frag(const float* __restrict__ Elds,
                                          const float* __restrict__ EsqLds,
                                          int j, int l, int kh) {
    Frag f;
    const int code = j * 16 + l;
    const float* er = &Elds[code * DPAD];
    f.b0.x = er[0 + kh * 2]; f.b0.y = er[1 + kh * 2];
    f.b1.x = er[4 + kh * 2]; f.b1.y = er[5 + kh * 2];
    f.b2.x = er[8 + kh * 2]; f.b2.y = er[9 + kh * 2];
    f.es = EsqLds[code];
    return f;
}

__global__ __launch_bounds__(256) void vq_main(const float* __restrict__ X,
                                               const float* __restrict__ E,
                                               float* __restrict__ Out,
                                               float* __restrict__ partial) {
    __shared__ float Elds[K_CODES * DPAD];        // 48 KB, padded stride 12 (conflict-free)
    __shared__ float EsqLds[K_CODES];             // 4 KB
    __shared__ float Xlds[ROWS_PER_BLOCK * DPAD]; // 6 KB, padded stride 12
    __shared__ float lossWave[WAVES_PER_BLOCK];

    const int tid = threadIdx.x;
    const int rowBlock = blockIdx.x * ROWS_PER_BLOCK;

    // Stage codebook into LDS (zero-padded cols 10,11) and compute e_sq on the fly.
    for (int c = tid; c < K_CODES; c += 256) {
        float s = 0.f;
#pragma unroll
        for (int d = 0; d < DIM; ++d) {
            float v = E[c * DIM + d];
            Elds[c * DPAD + d] = v;
            s += v * v;
        }
        Elds[c * DPAD + 10] = 0.f;
        Elds[c * DPAD + 11] = 0.f;
        EsqLds[c] = s;
    }
    // Stage this block's 128 query rows, zero-padded.
    for (int r = tid; r < ROWS_PER_BLOCK; r += 256) {
#pragma unroll
        for (int d = 0; d < DIM; ++d)
            Xlds[r * DPAD + d] = X[(size_t)(rowBlock + r) * DIM + d];
        Xlds[r * DPAD + 10] = 0.f;
        Xlds[r * DPAD + 11] = 0.f;
    }
    __syncthreads();

    const int lane = tid & 31;
    const int wave = tid >> 5;
    const int l  = lane & 15;  // B/C column (code), A row (query)
    const int kh = lane >> 4;  // K-half selector for A/B fragments
    const int waveRow = wave * ROWS_PER_WAVE;

    // A fragments (16x4 f32): lane l = row M, VGPR pair = K = c*4 + kh*2 + {0,1}
    v2f a0, a1, a2;
    {
        const float* xr = &Xlds[(waveRow + l) * DPAD];
        a0.x = xr[0 + kh * 2]; a0.y = xr[1 + kh * 2];
        a1.x = xr[4 + kh * 2]; a1.y = xr[5 + kh * 2];
        a2.x = xr[8 + kh * 2]; a2.y = xr[9 + kh * 2]; // kh==1 reads the zero pad
    }

    float bestD[8];
    int   bestI[8];
#pragma unroll
    for (int r = 0; r < 8; ++r) { bestD[r] = 3.402823e38f; bestI[r] = 0; }

    // Software pipeline: fragments for tiles {j, j+1} in flight while {j+2, j+3}
    // are prefetched from LDS; two independent WMMA accumulator chains per step.
    Frag f0 = load_frag(Elds, EsqLds, 0, l, kh);
    Frag f1 = load_frag(Elds, EsqLds, 1, l, kh);

    for (int j = 0; j < TILES; j += 2) {
        Frag n0 = load_frag(Elds, EsqLds, (j + 2) & (TILES - 1), l, kh);
        Frag n1 = load_frag(Elds, EsqLds, (j + 3) & (TILES - 1), l, kh);

        v8f acc0 = {};
        v8f acc1 = {};
        acc0 = __builtin_amdgcn_wmma_f32_16x16x4_f32(false, a0, false, f0.b0,
                                                     (short)0, acc0, false, false);
        acc1 = __builtin_amdgcn_wmma_f32_16x16x4_f32(false, a0, false, f1.b0,
                                                     (short)0, acc1, false, false);
        acc0 = __builtin_amdgcn_wmma_f32_16x16x4_f32(false, a1, false, f0.b1,
                                                     (short)0, acc0, false, false);
        acc1 = __builtin_amdgcn_wmma_f32_16x16x4_f32(false, a1, false, f1.b1,
                                                     (short)0, acc1, false, false);
        acc0 = __builtin_amdgcn_wmma_f32_16x16x4_f32(false, a2, false, f0.b2,
                                                     (short)0, acc0, false, false);
        acc1 = __builtin_amdgcn_wmma_f32_16x16x4_f32(false, a2, false, f1.b2,
                                                     (short)0, acc1, false, false);

        const int code0 = j * 16 + l;
        const int code1 = code0 + 16;
#pragma unroll
        for (int r = 0; r < 8; ++r) {
            float d0 = __builtin_fmaf(-2.f, acc0[r], f0.es);
            if (d0 < bestD[r]) { bestD[r] = d0; bestI[r] = code0; }
            float d1 = __builtin_fmaf(-2.f, acc1[r], f1.es);
            if (d1 < bestD[r]) { bestD[r] = d1; bestI[r] = code1; }
        }

        f0 = n0;
        f1 = n1;
    }

    // Reduce over the 16 lanes of each half-wave (codes), tie-break to lowest index.
#pragma unroll
    for (int m = 8; m; m >>= 1) {
#pragma unroll
        for (int r = 0; r < 8; ++r) {
            float od = __shfl_xor(bestD[r], m, 32);
            int   oi = __shfl_xor(bestI[r], m, 32);
            if (od < bestD[r] || (od == bestD[r] && oi < bestI[r])) {
                bestD[r] = od; bestI[r] = oi;
            }
        }
    }

    // Write gathered codewords + accumulate squared error. Lane l covers column l (<10)
    // of rows M = kh*8 + r of this wave's tile.
    float lacc = 0.f;
#pragma unroll
    for (int r = 0; r < 8; ++r) {
        const int localRow = waveRow + kh * 8 + r;
        const int row = rowBlock + localRow;
        const int code = bestI[r];
        if (l < DIM) {
            float e  = Elds[code * DPAD + l];
            float xv = Xlds[localRow * DPAD + l];
            Out[(size_t)row * DIM + l] = e;
            float df = xv - e;
            lacc += df * df;
        }
    }

    // Deterministic loss reduction: wave shuffle -> LDS -> block partial.
#pragma unroll
    for (int m = 16; m; m >>= 1) lacc += __shfl_xor(lacc, m, 32);
    if (lane == 0) lossWave[wave] = lacc;
    __syncthreads();
    if (tid == 0) {
        float s = 0.f;
#pragma unroll
        for (int w = 0; w < WAVES_PER_BLOCK; ++w) s += lossWave[w];
        partial[blockIdx.x] = s;
    }
}

__global__ __launch_bounds__(256) void vq_finalize(const float* __restrict__ partial,
                                                   float* __restrict__ out_loss) {
    __shared__ float sm[256];
    float s = 0.f;
    for (int i = threadIdx.x; i < NUM_BLOCKS; i += 256) s += partial[i];
    sm[threadIdx.x] = s;
    __syncthreads();
    for (int st = 128; st; st >>= 1) {
        if ((int)threadIdx.x < st) sm[threadIdx.x] += sm[threadIdx.x + st];
        __syncthreads();
    }
    if (threadIdx.x == 0) out_loss[0] = sm[0] / (float)(N_ROWS * DIM);
}

extern "C" void kernel_launch(void* const* d_in, const int* in_sizes, int n_in,
                              void* d_out, int out_size, void* d_ws, size_t ws_size,
                              hipStream_t stream) {
    (void)in_sizes; (void)n_in; (void)out_size; (void)ws_size;
    const float* X = (const float*)d_in[0]; // encoder_embedding [N, D]
    const float* E = (const float*)d_in[1]; // embedding_weight  [K, D]
    float* Out = (float*)d_out;             // [N*D] quantized_st, then [1] loss
    float* partial = (float*)d_ws;          // NUM_BLOCKS floats

    vq_main<<<NUM_BLOCKS, 256, 0, stream>>>(X, E, Out, partial);
    vq_finalize<<<1, 256, 0, stream>>>(partial, Out + (size_t)N_ROWS * DIM);
}